// Architecture_51161650430159
// MI455X (gfx1250) — compile-verified
//
#include <hip/hip_runtime.h>

#define HD    8
#define C2    16          // 2*HD
#define IMG   28
#define HW    784         // 28*28
#define BATCH 16
#define KDIM  6272        // HD*HW
#define NDIM  12544       // 2*HD*HW
#define SPLITK 4
#define KCH   (KDIM / SPLITK)   // 1568
#define NTILES (NDIM / 16)      // 784

typedef float v2f __attribute__((ext_vector_type(2)));
typedef float v8f __attribute__((ext_vector_type(8)));

__device__ __forceinline__ float sigmoidf(float x) { return 1.0f / (1.0f + __expf(-x)); }

// ---------------------------------------------------------------------------
// input conv: (B,3,28,28) -> (B,8,28,28), 3x3 SAME
// ---------------------------------------------------------------------------
__global__ void k_input_conv(const float* __restrict__ x, const float* __restrict__ w,
                             const float* __restrict__ bias, float* __restrict__ out) {
  int i = blockIdx.x * blockDim.x + threadIdx.x;
  if (i >= BATCH * HD * HW) return;
  int p = i % HW, oc = (i / HW) % HD, b = i / (HW * HD);
  int y = p / IMG, xx = p % IMG;
  float acc = bias[oc];
  for (int ic = 0; ic < 3; ++ic)
    for (int ky = 0; ky < 3; ++ky) {
      int yy = y + ky - 1; if ((unsigned)yy >= IMG) continue;
      for (int kx = 0; kx < 3; ++kx) {
        int xc = xx + kx - 1; if ((unsigned)xc >= IMG) continue;
        acc += w[((oc * 3 + ic) * 3 + ky) * 3 + kx] * x[(b * 3 + ic) * HW + yy * IMG + xc];
      }
    }
  out[i] = acc;
}

// td map = bias broadcast (sweep 0: td_proj input is exactly zero)
__global__ void k_bcast_bias(const float* __restrict__ bias, float* __restrict__ td) {
  int i = blockIdx.x * blockDim.x + threadIdx.x;
  if (i >= BATCH * NDIM) return;
  td[i] = bias[i % NDIM];
}

// Xs = s * h   (flat (B,6272))
__global__ void k_scale(const float* __restrict__ h, float s, float* __restrict__ xs) {
  int i = blockIdx.x * blockDim.x + threadIdx.x;
  if (i >= BATCH * KDIM) return;
  xs[i] = s * h[i];
}

// ---------------------------------------------------------------------------
// GRU phase 1: cm[b,c,p] = (c<8 ? xs*x : h) * sigmoid(td);  rhx[b,c<8,p] = xs*x
// ---------------------------------------------------------------------------
__global__ void k_gru_prep(const float* __restrict__ x, float xs,
                           const float* __restrict__ h, const float* __restrict__ td,
                           float* __restrict__ cm, float* __restrict__ rhx) {
  int i = blockIdx.x * blockDim.x + threadIdx.x;
  if (i >= BATCH * C2 * HW) return;
  int p = i % HW, c = (i / HW) % C2, b = i / (HW * C2);
  float s = sigmoidf(td[i]);
  float v;
  if (c < HD) { float xv = xs * x[(b * HD + c) * HW + p]; v = xv; rhx[i] = xv; }
  else        { v = h[(b * HD + (c - HD)) * HW + p]; }
  cm[i] = v * s;
}

// ---------------------------------------------------------------------------
// GRU phase 2: gates conv (16->16, 3x3) over cm; r -> rhx[8..15]=r*h, u -> u buf
// ---------------------------------------------------------------------------
__global__ void k_gru_gates(const float* __restrict__ cm, const float* __restrict__ h,
                            const float* __restrict__ gw, const float* __restrict__ gb,
                            float* __restrict__ rhx, float* __restrict__ u) {
  __shared__ float sw[C2 * C2 * 9];
  for (int t = threadIdx.x; t < C2 * C2 * 9; t += blockDim.x) sw[t] = gw[t];
  __syncthreads();
  int i = blockIdx.x * blockDim.x + threadIdx.x;
  if (i >= BATCH * C2 * HW) return;
  int p = i % HW, oc = (i / HW) % C2, b = i / (HW * C2);
  int y = p / IMG, xx = p % IMG;
  float acc = gb[oc];
  const float* base = cm + b * C2 * HW;
  for (int ic = 0; ic < C2; ++ic) {
    const float* wr = sw + (oc * C2 + ic) * 9;
    const float* cr = base + ic * HW;
    for (int ky = 0; ky < 3; ++ky) {
      int yy = y + ky - 1; if ((unsigned)yy >= IMG) continue;
      for (int kx = 0; kx < 3; ++kx) {
        int xc = xx + kx - 1; if ((unsigned)xc >= IMG) continue;
        acc += wr[ky * 3 + kx] * cr[yy * IMG + xc];
      }
    }
  }
  float g = sigmoidf(acc);
  if (oc < HD) rhx[(b * C2 + HD + oc) * HW + p] = g * h[(b * HD + oc) * HW + p];
  else         u[(b * HD + (oc - HD)) * HW + p] = g;
}

// ---------------------------------------------------------------------------
// GRU phase 3: cand conv (16->8, 3x3) over rhx; out = (1-u)*h + u*tanh(cand)
// out may alias h (elementwise; rhx/u already materialized)
// ---------------------------------------------------------------------------
__global__ void k_gru_cand(const float* __restrict__ rhx, const float* __restrict__ h,
                           const float* __restrict__ cw, const float* __restrict__ cb,
                           const float* __restrict__ u, float* __restrict__ out) {
  __shared__ float sw[HD * C2 * 9];
  for (int t = threadIdx.x; t < HD * C2 * 9; t += blockDim.x) sw[t] = cw[t];
  __syncthreads();
  int i = blockIdx.x * blockDim.x + threadIdx.x;
  if (i >= BATCH * HD * HW) return;
  int p = i % HW, hc = (i / HW) % HD, b = i / (HW * HD);
  int y = p / IMG, xx = p % IMG;
  float acc = cb[hc];
  const float* base = rhx + b * C2 * HW;
  for (int ic = 0; ic < C2; ++ic) {
    const float* wr = sw + (hc * C2 + ic) * 9;
    const float* cr = base + ic * HW;
    for (int ky = 0; ky < 3; ++ky) {
      int yy = y + ky - 1; if ((unsigned)yy >= IMG) continue;
      for (int kx = 0; kx < 3; ++kx) {
        int xc = xx + kx - 1; if ((unsigned)xc >= IMG) continue;
        acc += wr[ky * 3 + kx] * cr[yy * IMG + xc];
      }
    }
  }
  float cand = tanhf(acc);
  float uu = u[i];
  float hp = h[i];
  out[i] = (1.0f - uu) * hp + uu * cand;
}

// ---------------------------------------------------------------------------
// Big matmul: Ypart[kc] = Xs(16xKCH chunk) @ W(chunk)^T using V_WMMA_F32_16X16X4_F32
// One wave per (N-tile, K-chunk). A/B fragments per ISA 7.12.2:
//   A 16x4 f32: VGPR0 = {K=0 | K=2}, VGPR1 = {K=1 | K=3} -> b64 load @ k0+2*(lane>>4)
//   B 4x16 f32: same row-pairing pattern, n = nb + (lane&15)
//   C/D 16x16 f32: VGPR j -> M = j + 8*(lane>>4), N = nb + (lane&15)
// ---------------------------------------------------------------------------
__global__ void k_td_matmul(const float* __restrict__ Xs, const float* __restrict__ W,
                            float* __restrict__ Ypart) {
  int wid  = (blockIdx.x * blockDim.x + threadIdx.x) >> 5;
  int lane = threadIdx.x & 31;
  int tile = wid % NTILES;
  int kc   = wid / NTILES;
  if (kc >= SPLITK) return;
  int nb = tile * 16;
  int r  = lane & 15;
  int kh = (lane >> 4) * 2;

  const float* xrow = Xs + (size_t)r * KDIM + (size_t)kc * KCH + kh;
  const float* wrow = W + (size_t)(nb + r) * KDIM + (size_t)kc * KCH + kh;

  v8f acc = {};
#pragma unroll 8
  for (int k = 0; k < KCH; k += 4) {
    v2f a = *(const v2f*)(xrow + k);
    v2f bfrag = __builtin_nontemporal_load((const v2f*)(wrow + k));  // single-use weights
    acc = __builtin_amdgcn_wmma_f32_16x16x4_f32(false, a, false, bfrag,
                                                (short)0, acc, false, false);
  }

  float* yp = Ypart + (size_t)kc * (BATCH * NDIM);
  int mb  = (lane >> 4) * 8;
  int col = nb + r;
#pragma unroll
  for (int j = 0; j < 8; ++j)
    yp[(size_t)(mb + j) * NDIM + col] = acc[j];
}

// deterministic split-K reduction + bias
__global__ void k_td_reduce(const float* __restrict__ Ypart, const float* __restrict__ bias,
                            float* __restrict__ td) {
  int i = blockIdx.x * blockDim.x + threadIdx.x;
  if (i >= BATCH * NDIM) return;
  float s = bias[i % NDIM];
#pragma unroll
  for (int c = 0; c < SPLITK; ++c) s += Ypart[(size_t)c * (BATCH * NDIM) + i];
  td[i] = s;
}

// ---------------------------------------------------------------------------
// FC head
// ---------------------------------------------------------------------------
__global__ void k_fc1(const float* __restrict__ h2, const float* __restrict__ w,
                      const float* __restrict__ b, float* __restrict__ out) {
  int i = blockIdx.x * blockDim.x + threadIdx.x;
  if (i >= BATCH * 100) return;
  int o = i % 100, bb = i / 100;
  const float* hr = h2 + (size_t)bb * KDIM;
  const float* wr = w + (size_t)o * KDIM;
  float acc = b[o];
  for (int k = 0; k < KDIM; ++k) acc += fmaxf(hr[k], 0.0f) * wr[k];
  out[i] = acc;
}

__global__ void k_fc2(const float* __restrict__ f1, const float* __restrict__ w,
                      const float* __restrict__ b, float* __restrict__ out) {
  int i = blockIdx.x * blockDim.x + threadIdx.x;
  if (i >= BATCH * 10) return;
  int o = i % 10, bb = i / 10;
  const float* hr = f1 + bb * 100;
  const float* wr = w + o * 100;
  float acc = b[o];
  for (int k = 0; k < 100; ++k) acc += fmaxf(hr[k], 0.0f) * wr[k];
  out[i] = acc;
}

// ---------------------------------------------------------------------------

extern "C" void kernel_launch(void* const* d_in, const int* in_sizes, int n_in,
                              void* d_out, int out_size, void* d_ws, size_t ws_size,
                              hipStream_t stream) {
  const float* input_tensor = (const float*)d_in[0];
  const float* topdown      = (const float*)d_in[1];
  const float* in_conv_w    = (const float*)d_in[2];
  const float* in_conv_b    = (const float*)d_in[3];
  const float* gates_w      = (const float*)d_in[4];   // (3,16,16,3,3)
  const float* gates_b      = (const float*)d_in[5];   // (3,16)
  const float* can_w        = (const float*)d_in[6];   // (3,8,16,3,3)
  const float* can_b        = (const float*)d_in[7];   // (3,8)
  const float* td_w0        = (const float*)d_in[8];
  const float* td_b0        = (const float*)d_in[9];
  const float* td_w1        = (const float*)d_in[10];
  const float* td_b1        = (const float*)d_in[11];
  const float* fc1_w        = (const float*)d_in[12];
  const float* fc1_b        = (const float*)d_in[13];
  const float* fc2_w        = (const float*)d_in[14];
  const float* fc2_b        = (const float*)d_in[15];

  float* ws = (float*)d_ws;
  size_t o = 0;
  float* zero   = ws + o; o += BATCH * HD * HW;
  float* h0     = ws + o; o += BATCH * HD * HW;
  float* h1     = ws + o; o += BATCH * HD * HW;
  float* h2     = ws + o; o += BATCH * HD * HW;
  float* curin  = ws + o; o += BATCH * HD * HW;
  float* tdbuf  = ws + o; o += BATCH * C2 * HW;
  float* cm     = ws + o; o += BATCH * C2 * HW;
  float* rhx    = ws + o; o += BATCH * C2 * HW;
  float* ubuf   = ws + o; o += BATCH * HD * HW;
  float* Xs     = ws + o; o += BATCH * KDIM;
  float* Ypart  = ws + o; o += (size_t)SPLITK * BATCH * NDIM;
  float* fc1out = ws + o; o += BATCH * 100;

  hipMemsetAsync(zero, 0, (size_t)BATCH * HD * HW * sizeof(float), stream);

  const int N1 = BATCH * C2 * HW;   // 200704
  const int N2 = BATCH * HD * HW;   // 100352
  const int GSZ = C2 * C2 * 9, CSZ = HD * C2 * 9;

  auto gru = [&](const float* x, float xsc, const float* hin, float* hout,
                 const float* td, int node) {
    k_gru_prep <<<(N1 + 255) / 256, 256, 0, stream>>>(x, xsc, hin, td, cm, rhx);
    k_gru_gates<<<(N1 + 255) / 256, 256, 0, stream>>>(cm, hin, gates_w + node * GSZ,
                                                      gates_b + node * C2, rhx, ubuf);
    k_gru_cand <<<(N2 + 255) / 256, 256, 0, stream>>>(rhx, hin, can_w + node * CSZ,
                                                      can_b + node * HD, ubuf, hout);
  };

  // cur_input = conv(input_tensor[:, :, 0])
  k_input_conv<<<(N2 + 255) / 256, 256, 0, stream>>>(input_tensor, in_conv_w, in_conv_b, curin);

  // ---- sweep 0: prev = zeros, so td_proj == bias broadcast; x/h of nodes 1,2 zero
  k_bcast_bias<<<(N1 + 255) / 256, 256, 0, stream>>>(td_b0, tdbuf);
  gru(curin, 1.0f, zero, h0, tdbuf, 0);
  k_bcast_bias<<<(N1 + 255) / 256, 256, 0, stream>>>(td_b1, tdbuf);
  gru(zero, 0.8f, zero, h1, tdbuf, 1);
  gru(zero, 0.7f, zero, h2, topdown, 2);

  // ---- sweep 1: prev aliases cur -> in-sweep updates visible
  const int mmThreads = SPLITK * NTILES * 32;  // 100352
  // td0 = (0.6*h1) @ td_w0^T + b0
  k_scale    <<<(BATCH * KDIM + 255) / 256, 256, 0, stream>>>(h1, 0.6f, Xs);
  k_td_matmul<<<mmThreads / 256, 256, 0, stream>>>(Xs, td_w0, Ypart);
  k_td_reduce<<<(N1 + 255) / 256, 256, 0, stream>>>(Ypart, td_b0, tdbuf);
  gru(curin, 1.0f, h0, h0, tdbuf, 0);

  // td1 = (0.5*h2) @ td_w1^T + b1
  k_scale    <<<(BATCH * KDIM + 255) / 256, 256, 0, stream>>>(h2, 0.5f, Xs);
  k_td_matmul<<<mmThreads / 256, 256, 0, stream>>>(Xs, td_w1, Ypart);
  k_td_reduce<<<(N1 + 255) / 256, 256, 0, stream>>>(Ypart, td_b1, tdbuf);
  gru(h0, 0.8f, h1, h1, tdbuf, 1);          // x = 0.8 * updated node0
  gru(h1, 0.7f, h2, h2, topdown, 2);        // x = 0.7 * updated node1

  // ---- head
  k_fc1<<<(BATCH * 100 + 255) / 256, 256, 0, stream>>>(h2, fc1_w, fc1_b, fc1out);
  k_fc2<<<1, 256, 0, stream>>>(fc1out, fc2_w, fc2_b, (float*)d_out);
}